// Attention_68564857913412
// MI455X (gfx1250) — compile-verified
//
#include <hip/hip_runtime.h>
#include <math.h>

// Problem constants (from reference)
#define BB     64
#define TT     1024
#define RR     128
#define QD     1024
#define MD     512
#define CC     32
#define KK     31
#define PADK   15
#define TSPLIT 8                 // T-chunks for the context GEMV
#define TCHUNK (TT / TSPLIT)     // 128

typedef __attribute__((ext_vector_type(2))) float v2f;
typedef __attribute__((ext_vector_type(8))) float v8f;

__device__ __forceinline__ v8f wmma_f32_16x16x4(v2f a, v2f b, v8f c) {
  // 8 args: (neg_a, A, neg_b, B, c_mod, C, reuse_a, reuse_b)
  return __builtin_amdgcn_wmma_f32_16x16x4_f32(false, a, false, b, (short)0, c,
                                               false, false);
}

// ---------------------------------------------------------------------------
// Kernel A: Conv1d(1->C, K=31, pad=15) over cum_weights -> loc[B,T,C]
// lane = channel c; (tid>>5) selects one of 8 t's per block.
// ---------------------------------------------------------------------------
__global__ void __launch_bounds__(256) conv_loc_kernel(
    const float* __restrict__ cw,      // [B,T]
    const float* __restrict__ convw,   // [C,1,K]
    float* __restrict__ loc) {         // [B,T,C]
  const int c  = threadIdx.x & 31;
  const int bt = blockIdx.x * 8 + (threadIdx.x >> 5);
  const int b  = bt >> 10;             // T = 1024
  const int t  = bt & (TT - 1);
  float s = 0.f;
#pragma unroll
  for (int k = 0; k < KK; ++k) {
    const int tt = t + k - PADK;
    const float x = (tt >= 0 && tt < TT) ? cw[b * TT + tt] : 0.f;
    s += x * convw[c * KK + k];
  }
  loc[(size_t)bt * CC + c] = s;
}

// ---------------------------------------------------------------------------
// Kernel B: qb[B,R] = query @ Wq^T + bias, via V_WMMA_F32_16X16X4_F32.
// One wave per 16x16 output tile; K-loop of QD/4 = 256 WMMAs.
// A layout (16x4 f32): lanes 0-15 -> M=lane, {K=0,K=1}; lanes 16-31 -> {K=2,K=3}
// B layout (4x16 f32): lane -> N, half selects K pair (mirror of C/D striping).
// ---------------------------------------------------------------------------
__global__ void __launch_bounds__(32) qproj_wmma_kernel(
    const float* __restrict__ query,   // [B,QD]
    const float* __restrict__ Wq,      // [R,QD]
    const float* __restrict__ bias,    // [1,R]
    float* __restrict__ qb) {          // [B,R]
  const int lane = threadIdx.x;
  const int m0   = (blockIdx.x >> 3) * 16;  // B/16 = 4 tiles
  const int n0   = (blockIdx.x & 7) * 16;   // R/16 = 8 tiles
  const int half = lane >> 4;               // 0: K lo pair, 1: K hi pair
  const int row  = lane & 15;               // M for A, N for B/C/D

  v8f acc = {};
  for (int k0 = 0; k0 < QD; k0 += 4) {
    const int kh = k0 + half * 2;
    v2f a, w;
    a.x = query[(size_t)(m0 + row) * QD + kh];
    a.y = query[(size_t)(m0 + row) * QD + kh + 1];
    // B[k][n] = Wq^T[k][n] = Wq[n][k]
    w.x = Wq[(size_t)(n0 + row) * QD + kh];
    w.y = Wq[(size_t)(n0 + row) * QD + kh + 1];
    acc = wmma_f32_16x16x4(a, w, acc);
  }
#pragma unroll
  for (int v = 0; v < 8; ++v) {
    const int M = v + half * 8;
    const int N = n0 + row;
    qb[(size_t)(m0 + M) * RR + N] = acc[v] + bias[N];
  }
}

// ---------------------------------------------------------------------------
// Kernel C: 8 waves/block, one wave per (b, 16-row t-tile):
//   loc_feat = loc_tile(16x32) @ Wloc^T(32xR) via WMMA (8 K-steps x 8 r-tiles),
//   energy = tanh(qb + memory_transform + loc_feat),
//   energies[b,t] = sum_r energy * w_energy[r], masked.
// memory_transform streamed once, loaded directly in C/D VGPR layout.
// ---------------------------------------------------------------------------
__global__ void __launch_bounds__(256) energy_wmma_kernel(
    const float* __restrict__ loc,     // [B,T,C]
    const float* __restrict__ Wloc,    // [R,C]
    const float* __restrict__ mt,      // [B,T,R]
    const float* __restrict__ qb,      // [B,R] (bias folded in)
    const float* __restrict__ wE,      // [1,R]
    const unsigned char* __restrict__ mask,  // [B,T] bool
    float* __restrict__ energ) {       // [B,T]
  const int lane = threadIdx.x & 31;
  const int tile = blockIdx.x * 8 + (threadIdx.x >> 5);  // BB*T/16 tiles total
  const int b    = tile >> 6;                // T/16 = 64 tiles per b
  const int t0   = (tile & 63) * 16;
  const int half = lane >> 4;
  const int row  = lane & 15;

  // Hoist full A tile: loc[t0..t0+15][0..31] -> 8 x v2f (16 VGPRs)
  v2f a[8];
#pragma unroll
  for (int kc = 0; kc < 8; ++kc) {
    const int kk = kc * 4 + half * 2;
    const float* p = loc + ((size_t)b * TT + t0 + row) * CC + kk;
    a[kc].x = p[0];
    a[kc].y = p[1];
  }

  float esum[8];
#pragma unroll
  for (int v = 0; v < 8; ++v) esum[v] = 0.f;

  for (int r0 = 0; r0 < RR; r0 += 16) {
    v8f acc = {};
#pragma unroll
    for (int kc = 0; kc < 8; ++kc) {
      const int kk = kc * 4 + half * 2;
      v2f w;  // B[k][n] = Wloc^T[k][n] = Wloc[n][k]
      w.x = Wloc[(size_t)(r0 + row) * CC + kk];
      w.y = Wloc[(size_t)(r0 + row) * CC + kk + 1];
      acc = wmma_f32_16x16x4(a[kc], w, acc);
    }
    const int   N   = r0 + row;
    const float qbe = qb[(size_t)b * RR + N];
    const float we  = wE[N];
#pragma unroll
    for (int v = 0; v < 8; ++v) {
      const int t = t0 + v + half * 8;
      const float e =
          tanhf(acc[v] + mt[((size_t)b * TT + t) * RR + N] + qbe);
      esum[v] += e * we;
    }
  }

  // Reduce over the 16 N-lanes inside each half-wave (wave32, width=16).
#pragma unroll
  for (int v = 0; v < 8; ++v) {
    float s = esum[v];
    s += __shfl_xor(s, 8, 16);
    s += __shfl_xor(s, 4, 16);
    s += __shfl_xor(s, 2, 16);
    s += __shfl_xor(s, 1, 16);
    if (row == 0) {
      const int t = t0 + v + half * 8;
      const float val = mask[(size_t)b * TT + t] ? s : -INFINITY;
      energ[(size_t)b * TT + t] = val;
    }
  }
}

// ---------------------------------------------------------------------------
// Kernel D: masked softmax over T per batch row. 256 threads, 4 elems each.
// ---------------------------------------------------------------------------
__global__ void __launch_bounds__(256) softmax_kernel(
    const float* __restrict__ energ,   // [B,T]
    float* __restrict__ attn) {        // [B,T] (written into d_out attn slot)
  __shared__ float red[256];
  const int b = blockIdx.x, tid = threadIdx.x;
  float e[4];
  float m = -INFINITY;
#pragma unroll
  for (int i = 0; i < 4; ++i) {
    e[i] = energ[(size_t)b * TT + tid + i * 256];
    m = fmaxf(m, e[i]);
  }
  red[tid] = m;
  __syncthreads();
  for (int s = 128; s > 0; s >>= 1) {
    if (tid < s) red[tid] = fmaxf(red[tid], red[tid + s]);
    __syncthreads();
  }
  m = red[0];
  __syncthreads();
  float sum = 0.f;
#pragma unroll
  for (int i = 0; i < 4; ++i) {
    e[i] = __expf(e[i] - m);
    sum += e[i];
  }
  red[tid] = sum;
  __syncthreads();
  for (int s = 128; s > 0; s >>= 1) {
    if (tid < s) red[tid] += red[tid + s];
    __syncthreads();
  }
  const float inv = 1.0f / red[0];
#pragma unroll
  for (int i = 0; i < 4; ++i)
    attn[(size_t)b * TT + tid + i * 256] = e[i] * inv;
}

// ---------------------------------------------------------------------------
// Kernel E1: partial context over a T-chunk.
//   partial[chunk][b][d] = sum_{t in chunk} attn[b,t] * memory[b,t,d]
// 512 blocks x 256 threads streaming the 134 MB memory tensor with float2
// loads; x4 unroll keeps multiple global_load_b64 in flight per thread.
// ---------------------------------------------------------------------------
__global__ void __launch_bounds__(256) context_partial_kernel(
    const float* __restrict__ attn,    // [B,T]
    const float* __restrict__ memory,  // [B,T,MD]
    float* __restrict__ partial) {     // [TSPLIT,B,MD]
  __shared__ float sa[TCHUNK];
  const int tid   = threadIdx.x;
  const int b     = blockIdx.x / TSPLIT;
  const int chunk = blockIdx.x % TSPLIT;
  const int t0    = chunk * TCHUNK;
  if (tid < TCHUNK) sa[tid] = attn[(size_t)b * TT + t0 + tid];
  __syncthreads();

  const float2* mem = (const float2*)(memory + ((size_t)b * TT + t0) * MD);
  float2 acc = make_float2(0.f, 0.f);
#pragma unroll 4
  for (int t = 0; t < TCHUNK; ++t) {
    const float a = sa[t];
    const float2 v = mem[(size_t)t * (MD / 2) + tid];
    acc.x += a * v.x;
    acc.y += a * v.y;
  }
  ((float2*)(partial + ((size_t)chunk * BB + b) * MD))[tid] = acc;
}

// ---------------------------------------------------------------------------
// Kernel E2: deterministic reduction of the TSPLIT partial contexts.
// ---------------------------------------------------------------------------
__global__ void __launch_bounds__(256) context_reduce_kernel(
    const float* __restrict__ partial, // [TSPLIT,B,MD]
    float* __restrict__ ctx) {         // [B,MD]
  const int b = blockIdx.x, tid = threadIdx.x;
  float2 acc = make_float2(0.f, 0.f);
#pragma unroll
  for (int c = 0; c < TSPLIT; ++c) {
    const float2 v =
        ((const float2*)(partial + ((size_t)c * BB + b) * MD))[tid];
    acc.x += v.x;
    acc.y += v.y;
  }
  ((float2*)(ctx + (size_t)b * MD))[tid] = acc;
}

// ---------------------------------------------------------------------------
extern "C" void kernel_launch(void* const* d_in, const int* in_sizes, int n_in,
                              void* d_out, int out_size, void* d_ws,
                              size_t ws_size, hipStream_t stream) {
  const float*         query  = (const float*)d_in[0];         // [B,QD]
  const float*         memory = (const float*)d_in[1];         // [B,T,MD]
  const float*         mt     = (const float*)d_in[2];         // [B,T,R]
  const float*         cw     = (const float*)d_in[3];         // [B,T]
  const unsigned char* mask   = (const unsigned char*)d_in[4]; // [B,T] bool
  const float*         Wq     = (const float*)d_in[5];         // [R,QD]
  const float*         Wloc   = (const float*)d_in[6];         // [R,C]
  const float*         convw  = (const float*)d_in[7];         // [C,1,K]
  const float*         bias   = (const float*)d_in[8];         // [1,R]
  const float*         wE     = (const float*)d_in[9];         // [1,R]

  // Workspace (floats): loc[B*T*C] | qb[B*R] | energ[B*T] | partial[TSPLIT*B*MD]
  float* loc     = (float*)d_ws;
  float* qb      = loc + (size_t)BB * TT * CC;
  float* energ   = qb + (size_t)BB * RR;
  float* partial = energ + (size_t)BB * TT;

  // Output layout: context[B*MD] then attn[B*1*T]
  float* ctx      = (float*)d_out;
  float* attn_out = ctx + (size_t)BB * MD;

  conv_loc_kernel<<<BB * TT / 8, 256, 0, stream>>>(cw, convw, loc);
  qproj_wmma_kernel<<<(BB / 16) * (RR / 16), 32, 0, stream>>>(query, Wq, bias,
                                                              qb);
  energy_wmma_kernel<<<BB * (TT / 16) / 8, 256, 0, stream>>>(loc, Wloc, mt, qb,
                                                             wE, mask, energ);
  softmax_kernel<<<BB, 256, 0, stream>>>(energ, attn_out);
  context_partial_kernel<<<BB * TSPLIT, 256, 0, stream>>>(attn_out, memory,
                                                          partial);
  context_reduce_kernel<<<BB, 256, 0, stream>>>(partial, ctx);
}